// NeRFDecoderHead_49357764165847
// MI455X (gfx1250) — compile-verified
//
#include <hip/hip_runtime.h>
#include <math.h>

// NeRF depth renderer for MI455X (gfx1250, wave32).
// One ray per lane; sequential transmittance scan; trilinear gathers from an
// L2-resident 2.56MB grid (192MB L2 -> all gathers L2-hot); wave-level early
// exit via ballot_w32 once every lane has left the box; global_prefetch_b8 of
// the next sample's x-slabs overlapped with the exp/log chain.

namespace {
constexpr int   kNumRays  = 6 * 90 * 160;   // 86400
constexpr int   kNSamples = 287;            // int(norm([101,101,9])/0.5)+1
constexpr float kZStep    = 0.2f;           // STEPSIZE * VOXEL_SIZE
constexpr float kInvZStep = 5.0f;
constexpr float kDelta    = 0.2f;           // zval[-1]-zval[-2]
constexpr int   kStrideX  = 200 * 16;       // 3200 floats
constexpr int   kStrideY  = 16;

// Branchless slab-exit time along one axis. inv = 1/d:
//  d==0, o inside  -> (-inf,+inf) -> fmax = +inf (axis never bounds)
//  o exactly on a face with d==0 -> 0*inf = NaN, fmaxf drops NaN -> +inf
__device__ __forceinline__ float axis_exit_t(float o, float d, float lo, float hi) {
  const float inv = __frcp_rn(d);
  return fmaxf((lo - o) * inv, (hi - o) * inv);
}
} // namespace

__global__ __launch_bounds__(256) void nerf_depth_kernel(
    const float* __restrict__ rays_o,
    const float* __restrict__ rays_d,
    const float* __restrict__ vox,
    float* __restrict__ out)
{
  const int ray   = blockIdx.x * 256 + threadIdx.x;
  const bool valid = (ray < kNumRays);

  float ox = 0.f, oy = 0.f, oz = 0.f, dx = 0.f, dy = 0.f, dz = 0.f;
  if (valid) {
    ox = rays_o[3 * ray + 0];
    oy = rays_o[3 * ray + 1];
    oz = rays_o[3 * ray + 2];
    dx = rays_d[3 * ray + 0];
    dy = rays_d[3 * ray + 1];
    dz = rays_d[3 * ray + 2];
  }

  // Index-space affine params: idx = (p - xyz_min) * (size-1)/(max-min)
  const float sx = 199.0f / 80.0f;   // x,y: (200-1)/(40-(-40))
  const float sy = 199.0f / 80.0f;
  const float sz = 15.0f / 6.4f;     // z:   (16-1)/(5.4-(-1))
  const float oix = (ox + 40.0f) * sx, dix = dx * sx;
  const float oiy = (oy + 40.0f) * sy, diy = dy * sy;
  const float oiz = (oz + 1.0f)  * sz, diz = dz * sz;

  // A line leaves the convex box exactly once (origins in [0,1]^3 start
  // inside). Margin covers FP slop; the exact in-loop mask preserves
  // reference semantics — t_exit only bounds the loop.
  float t_exit = fminf(fminf(axis_exit_t(ox, dx, -40.0f, 40.0f),
                             axis_exit_t(oy, dy, -40.0f, 40.0f)),
                             axis_exit_t(oz, dz,  -1.0f,  5.4f));
  t_exit = fminf(t_exit + 0.41f, 60.0f);            // clamp inf; z max = 57.2
  int s_max = (int)floorf(t_exit * kInvZStep);      // last sample to visit
  s_max = min(s_max, kNSamples - 1);
  if (!valid) s_max = -1;

  // Per-sample index-space step (for next-iteration prefetch).
  const float stx = dix * kZStep, sty = diy * kZStep, stz = diz * kZStep;

  float T = 1.0f;
  float depth = 0.0f;

  for (int s = 0; s < kNSamples; ++s) {
    const bool act = (s <= s_max);
    // Wave-level early termination once all 32 lanes exited the box.
    if (__builtin_amdgcn_ballot_w32(act) == 0u) break;

    if (act) {
      const float z  = kZStep * (float)s;
      const float tx = fmaf(dix, z, oix);
      const float ty = fmaf(diy, z, oiy);
      const float tz = fmaf(diz, z, oiz);
      const bool inb = (tx >= 0.0f) & (tx <= 199.0f) &
                       (ty >= 0.0f) & (ty <= 199.0f) &
                       (tz >= 0.0f) & (tz <= 15.0f);
      if (inb) {
        float fx0 = floorf(tx); fx0 = fminf(fx0, 198.0f);
        float fy0 = floorf(ty); fy0 = fminf(fy0, 198.0f);
        float fz0 = floorf(tz); fz0 = fminf(fz0, 14.0f);
        const int x0 = (int)fx0, y0 = (int)fy0, z0 = (int)fz0;
        const float fx = tx - fx0, fy = ty - fy0, fz = tz - fz0;

        const float* b = vox + (size_t)x0 * kStrideX + y0 * kStrideY + z0;

        // Prefetch next sample's two x-slabs (gfx1250 global_prefetch_b8);
        // warms WGP$ while this sample's exp/log chain executes.
        {
          const float txn = fminf(fmaxf(tx + stx, 0.0f), 198.0f);
          const float tyn = fminf(fmaxf(ty + sty, 0.0f), 198.0f);
          const float tzn = fminf(fmaxf(tz + stz, 0.0f), 14.0f);
          const float* pb = vox + (size_t)(int)txn * kStrideX
                                + (int)tyn * kStrideY + (int)tzn;
          __builtin_prefetch(pb, 0, 3);
          __builtin_prefetch(pb + kStrideX, 0, 3);
        }

        // 8 corner gathers; z-pairs are adjacent (z innermost, stride 1).
        const float c000 = b[0];
        const float c001 = b[1];
        const float c010 = b[kStrideY + 0];
        const float c011 = b[kStrideY + 1];
        const float c100 = b[kStrideX + 0];
        const float c101 = b[kStrideX + 1];
        const float c110 = b[kStrideX + kStrideY + 0];
        const float c111 = b[kStrideX + kStrideY + 1];

        // Trilinear interpolation as a nested-lerp FMA chain.
        const float c00 = fmaf(fz, c001 - c000, c000);
        const float c01 = fmaf(fz, c011 - c010, c010);
        const float c10 = fmaf(fz, c101 - c100, c100);
        const float c11 = fmaf(fz, c111 - c110, c110);
        const float c0  = fmaf(fy, c01 - c00, c00);
        const float c1  = fmaf(fy, c11 - c10, c10);
        const float dens = fmaf(fx, c1 - c0, c0);

        // alpha = 1 - exp(-softplus(dens) * delta), stable softplus.
        const float e  = __expf(-fabsf(dens));
        const float sp = fmaxf(dens, 0.0f) + __logf(1.0f + e);
        const float a  = 1.0f - __expf(-sp * kDelta);

        // Transmittance scan: weight = a * T_prev; T *= clip(1-a, 1e-10).
        depth = fmaf(a * T, z, depth);
        T *= fmaxf(1.0f - a, 1e-10f);
      }
      // Samples inside [0, s_max] but outside the box: alpha == 0 -> no-op.
    }
  }

  if (valid) out[ray] = depth;
}

extern "C" void kernel_launch(void* const* d_in, const int* in_sizes, int n_in,
                              void* d_out, int out_size, void* d_ws, size_t ws_size,
                              hipStream_t stream) {
  (void)in_sizes; (void)n_in; (void)d_ws; (void)ws_size; (void)out_size;
  const float* rays_o = (const float*)d_in[0];
  const float* rays_d = (const float*)d_in[1];
  const float* vox    = (const float*)d_in[2];
  float* out = (float*)d_out;

  const int threads = 256;                                 // 8 wave32 waves
  const int blocks  = (kNumRays + threads - 1) / threads;  // 338
  nerf_depth_kernel<<<blocks, threads, 0, stream>>>(rays_o, rays_d, vox, out);
}